// TangentInvariant_42039139894264
// MI455X (gfx1250) — compile-verified
//
#include <hip/hip_runtime.h>
#include <hip/hip_bf16.h>

typedef __attribute__((ext_vector_type(2))) float v2f;
typedef __attribute__((ext_vector_type(4))) float v4f;
typedef __attribute__((ext_vector_type(8))) float v8f;

#define NTOK 32            // tokens per workgroup (two WMMA M-tiles)
#define NIN 31             // C-1 tangent vectors
#define VO 16              // VEC_OUT
#define KDIM 496           // NIN * VO
#define KPAD 500           // padded LDS row
#define OUTC 128
#define NKP (KDIM / 8)     // 62 K-pair steps (2 WMMA k-steps each)
#define NEG_SLOPE 0.2f
#define EPS_F 2.2204460492503131e-16f   // np.finfo(float64).eps

// ---- one-time pack of dense_kernel into per-lane WMMA B-fragment order ----
// dkT[w][kp][lane][c] : c = {kk=2kp:K+0, kk=2kp:K+1, kk=2kp+1:K+0, kk=2kp+1:K+1}
// with K = 4*kk + khalf(lane), N = 16*w + (lane&15)
__global__ __launch_bounds__(256)
void pack_dense_kernel(const float* __restrict__ dk, float* __restrict__ dkT) {
    const int idx  = blockIdx.x * 256 + threadIdx.x;   // 0 .. 8*62*128-1 (exact)
    const int w    = idx / (NKP * 128);
    const int rem  = idx - w * (NKP * 128);
    const int kp   = rem >> 7;
    const int r2   = rem & 127;
    const int lane = r2 >> 2;
    const int c    = r2 & 3;
    const int kk   = 2 * kp + (c >> 1);
    const int j    = c & 1;
    const int khalf = (lane >> 4) << 1;
    const int ncol  = 16 * w + (lane & 15);
    dkT[idx] = dk[(size_t)(4 * kk + khalf + j) * OUTC + ncol];
}

template <bool PACKED>
__global__ __launch_bounds__(256)
void tangent_invariant_fused(const float* __restrict__ x,
                             const float* __restrict__ U0,
                             const float* __restrict__ W0,
                             const float* __restrict__ dkB,  // PACKED ? dkT : dk
                             const float* __restrict__ bias,
                             float* __restrict__ out)
{
    __shared__ float sV[NTOK][NIN * 3 + 1];  // tangent vectors v_i
    __shared__ float sW[NTOK][VO * 3];       // VN-layer outputs w_k
    __shared__ float sY[NTOK][KPAD];         // feature rows (A tiles, 32 x 496)

    const int tid = threadIdx.x;
    const int tokBase = blockIdx.x * NTOK;

    // ---------------- Phase 1: geometry -> 32x496 feature tile in LDS ----
    {
        const int tl = tid >> 3;   // local token 0..31
        const int r  = tid & 7;    // role within token 0..7
        const float* xt = x + (size_t)(tokBase + tl) * 32 * 3;

        const float px = xt[0], py = xt[1], pz = xt[2];

        // sphere log: role r handles tangent indices r, r+8, r+16, r+24
        for (int j = r; j < NIN; j += 8) {
            const float* q = xt + (j + 1) * 3;
            float qx = q[0], qy = q[1], qz = q[2];
            float c = px * qx + py * qy + pz * qz;
            c = fminf(1.0f, fmaxf(-1.0f, c));
            float th  = acosf(c);
            float fac = (th < 1e-6f) ? 1.0f : (th / sinf(th));
            sV[tl][j * 3 + 0] = fac * (qx - c * px);
            sV[tl][j * 3 + 1] = fac * (qy - c * py);
            sV[tl][j * 3 + 2] = fac * (qz - c * pz);
        }
        __syncthreads();

        // VN layer: role r computes output rows r and r+8
        float k0x = 0.f, k0y = 0.f, k0z = 0.f, q0x = 0.f, q0y = 0.f, q0z = 0.f;
        float k1x = 0.f, k1y = 0.f, k1z = 0.f, q1x = 0.f, q1y = 0.f, q1z = 0.f;
        #pragma unroll
        for (int j = 0; j < NIN; ++j) {
            float vx = sV[tl][j * 3 + 0];
            float vy = sV[tl][j * 3 + 1];
            float vz = sV[tl][j * 3 + 2];
            float u0 = U0[r * NIN + j],       w0 = W0[r * NIN + j];
            float u1 = U0[(r + 8) * NIN + j], w1 = W0[(r + 8) * NIN + j];
            k0x += u0 * vx; k0y += u0 * vy; k0z += u0 * vz;
            q0x += w0 * vx; q0y += w0 * vy; q0z += w0 * vz;
            k1x += u1 * vx; k1y += u1 * vy; k1z += u1 * vz;
            q1x += w1 * vx; q1y += w1 * vy; q1z += w1 * vz;
        }
        // w_out = q + (1-NEG_SLOPE) * k * relu(-q.k) / (|k|^2 + eps)
        {
            float sqn = k0x * k0x + k0y * k0y + k0z * k0z + EPS_F;
            float dqk = q0x * k0x + q0y * k0y + q0z * k0z;
            float t = (1.0f - NEG_SLOPE) * fmaxf(-dqk, 0.0f) / sqn;
            sW[tl][r * 3 + 0] = q0x + t * k0x;
            sW[tl][r * 3 + 1] = q0y + t * k0y;
            sW[tl][r * 3 + 2] = q0z + t * k0z;
        }
        {
            float sqn = k1x * k1x + k1y * k1y + k1z * k1z + EPS_F;
            float dqk = q1x * k1x + q1y * k1y + q1z * k1z;
            float t = (1.0f - NEG_SLOPE) * fmaxf(-dqk, 0.0f) / sqn;
            sW[tl][(r + 8) * 3 + 0] = q1x + t * k1x;
            sW[tl][(r + 8) * 3 + 1] = q1y + t * k1y;
            sW[tl][(r + 8) * 3 + 2] = q1z + t * k1z;
        }
        __syncthreads();

        // y[tl][i*16 + k_out] = v_i . w_{k_out}; role r fills columns r, r+8
        float w0x = sW[tl][r * 3 + 0],       w0y = sW[tl][r * 3 + 1],       w0z = sW[tl][r * 3 + 2];
        float w1x = sW[tl][(r + 8) * 3 + 0], w1y = sW[tl][(r + 8) * 3 + 1], w1z = sW[tl][(r + 8) * 3 + 2];
        #pragma unroll
        for (int i = 0; i < NIN; ++i) {
            float vx = sV[tl][i * 3 + 0];
            float vy = sV[tl][i * 3 + 1];
            float vz = sV[tl][i * 3 + 2];
            sY[tl][i * VO + r]     = vx * w0x + vy * w0y + vz * w0z;
            sY[tl][i * VO + r + 8] = vx * w1x + vy * w1y + vz * w1z;
        }
        __syncthreads();
    }

    // ------- Phase 2: WMMA f32 GEMM (2 x 16x496 tiles) @ (496x128) -------
    // wave w owns output columns [16w,16w+16); no divergence -> EXEC all 1s.
    const int lane  = tid & 31;
    const int wave  = tid >> 5;          // n-tile 0..7
    const int row   = lane & 15;         // M (A) / N (B, D)
    const int khalf = (lane >> 4) << 1;  // lanes 16-31 hold K={2,3} half
    const int ncol  = wave * 16 + row;

    v8f acc0 = {};   // tokens [0,16)
    v8f acc1 = {};   // tokens [16,32)
    #pragma unroll 2
    for (int kp = 0; kp < NKP; ++kp) {
        const int kb = kp * 8 + khalf;
        v2f b0, b1;
        if constexpr (PACKED) {
            // one coalesced b128: both K-step fragments for this lane
            const v4f bq = *(const v4f*)(dkB + (((size_t)wave * NKP + kp) * 32 + lane) * 4);
            b0.x = bq.x; b0.y = bq.y;
            b1.x = bq.z; b1.y = bq.w;
        } else {
            b0.x = dkB[(size_t)kb * OUTC + ncol];
            b0.y = dkB[(size_t)(kb + 1) * OUTC + ncol];
            b1.x = dkB[(size_t)(kb + 4) * OUTC + ncol];
            b1.y = dkB[(size_t)(kb + 5) * OUTC + ncol];
        }
        v2f a00, a01, a10, a11;      // {m-tile, k-step}
        a00.x = sY[row][kb];          a00.y = sY[row][kb + 1];
        a01.x = sY[row][kb + 4];      a01.y = sY[row][kb + 5];
        a10.x = sY[row + 16][kb];     a10.y = sY[row + 16][kb + 1];
        a11.x = sY[row + 16][kb + 4]; a11.y = sY[row + 16][kb + 5];
        // interleave the two independent accumulation chains
        acc0 = __builtin_amdgcn_wmma_f32_16x16x4_f32(false, a00, false, b0, (short)0, acc0, false, false);
        acc1 = __builtin_amdgcn_wmma_f32_16x16x4_f32(false, a10, false, b0, (short)0, acc1, false, false);
        acc0 = __builtin_amdgcn_wmma_f32_16x16x4_f32(false, a01, false, b1, (short)0, acc0, false, false);
        acc1 = __builtin_amdgcn_wmma_f32_16x16x4_f32(false, a11, false, b1, (short)0, acc1, false, false);
    }

    const float bb = bias[ncol];
    // C/D 16x16 f32 layout: VGPR r -> M=r (lanes 0-15) / M=r+8 (lanes 16-31)
    #pragma unroll
    for (int r = 0; r < 8; ++r) {
        const int M = r + ((lane >> 4) << 3);
        out[(size_t)(tokBase + M) * OUTC + ncol]        = acc0[r] + bb;
        out[(size_t)(tokBase + 16 + M) * OUTC + ncol]   = acc1[r] + bb;
    }
}

extern "C" void kernel_launch(void* const* d_in, const int* in_sizes, int n_in,
                              void* d_out, int out_size, void* d_ws, size_t ws_size,
                              hipStream_t stream) {
    const float* x    = (const float*)d_in[0];   // (16, 8192, 32, 3) f32
    const float* U0   = (const float*)d_in[1];   // (16, 31) f32
    const float* W0   = (const float*)d_in[2];   // (16, 31) f32
    const float* dk   = (const float*)d_in[3];   // (496, 128) f32
    const float* bias = (const float*)d_in[4];   // (128,) f32
    float* out = (float*)d_out;                  // (16, 8192, 128) f32

    const int ntok = in_sizes[0] / (32 * 3);     // 131072
    const int grid = ntok / NTOK;                // 4096 workgroups

    const size_t packElems = (size_t)8 * NKP * 128;          // 63488 floats
    const size_t packBytes = packElems * sizeof(float);      // 253,952 B

    if (ws_size >= packBytes) {
        float* dkT = (float*)d_ws;
        hipLaunchKernelGGL(pack_dense_kernel, dim3((int)(packElems / 256)), dim3(256),
                           0, stream, dk, dkT);
        hipLaunchKernelGGL(tangent_invariant_fused<true>, dim3(grid), dim3(256),
                           0, stream, x, U0, W0, dkT, bias, out);
    } else {
        hipLaunchKernelGGL(tangent_invariant_fused<false>, dim3(grid), dim3(256),
                           0, stream, x, U0, W0, dk, bias, out);
    }
}